// TransGCN_26345329394244
// MI455X (gfx1250) — compile-verified
//
#include <hip/hip_runtime.h>

typedef __attribute__((ext_vector_type(2))) float v2f;
typedef __attribute__((ext_vector_type(8))) float v8f;

#define NF 128   // NFEAT == NHID == 128

// ---------------------------------------------------------------- utilities
__global__ void zero_int_kernel(int* p, int n) {
  int i = blockIdx.x * blockDim.x + threadIdx.x;
  int stride = gridDim.x * blockDim.x;
  for (; i < n; i += stride) p[i] = 0;
}

// cnt_row[i] = out-degree by row (num_neighbor); deg_col[i] = in-degree by col
__global__ void degree_kernel(const int* __restrict__ row, const int* __restrict__ col,
                              int* __restrict__ cnt_row, int* __restrict__ deg_col, int E) {
  int i = blockIdx.x * blockDim.x + threadIdx.x;
  int stride = gridDim.x * blockDim.x;
  for (; i < E; i += stride) {
    atomicAdd(&cnt_row[row[i]], 1);
    atomicAdd(&deg_col[col[i]], 1);
  }
}

// Single-block exclusive prefix scan of two arrays (N up to a few 100k is fine).
__global__ void scan2_kernel(const int* __restrict__ a, const int* __restrict__ b,
                             int* __restrict__ oa, int* __restrict__ ob, int N) {
  __shared__ int sh[1024];
  const int T = 1024;
  int carryA = 0, carryB = 0;
  for (int base = 0; base < N; base += T) {
    int i = base + (int)threadIdx.x;
    // --- array a ---
    int va = (i < N) ? a[i] : 0;
    sh[threadIdx.x] = va;
    __syncthreads();
    for (int off = 1; off < T; off <<= 1) {
      int t = (threadIdx.x >= (unsigned)off) ? sh[threadIdx.x - off] : 0;
      __syncthreads();
      sh[threadIdx.x] += t;
      __syncthreads();
    }
    if (i < N) oa[i] = carryA + sh[threadIdx.x] - va;   // exclusive
    carryA += sh[T - 1];
    __syncthreads();
    // --- array b ---
    int vb = (i < N) ? b[i] : 0;
    sh[threadIdx.x] = vb;
    __syncthreads();
    for (int off = 1; off < T; off <<= 1) {
      int t = (threadIdx.x >= (unsigned)off) ? sh[threadIdx.x - off] : 0;
      __syncthreads();
      sh[threadIdx.x] += t;
      __syncthreads();
    }
    if (i < N) ob[i] = carryB + sh[threadIdx.x] - vb;
    carryB += sh[T - 1];
    __syncthreads();
  }
  if (threadIdx.x == 0) { oa[N] = carryA; ob[N] = carryB; }
}

__global__ void fill_cursor_kernel(const int* __restrict__ row_ofs, const int* __restrict__ col_ofs,
                                   int* __restrict__ cur_r, int* __restrict__ cur_c, int N) {
  int i = blockIdx.x * blockDim.x + threadIdx.x;
  int stride = gridDim.x * blockDim.x;
  for (; i < N; i += stride) { cur_r[i] = row_ofs[i]; cur_c[i] = col_ofs[i]; }
}

// Bucket edges: CSR by row (stores col), CSC by col (stores row). 2 int atomics/edge.
__global__ void build_adj_kernel(const int* __restrict__ row, const int* __restrict__ col,
                                 int* __restrict__ cur_r, int* __restrict__ cur_c,
                                 int* __restrict__ csr_col, int* __restrict__ csc_row, int E) {
  int i = blockIdx.x * blockDim.x + threadIdx.x;
  int stride = gridDim.x * blockDim.x;
  for (; i < E; i += stride) {
    int r = row[i], c = col[i];
    int p = atomicAdd(&cur_r[r], 1);
    csr_col[p] = c;
    int q = atomicAdd(&cur_c[c], 1);
    csc_row[q] = r;
  }
}

// invnb = 1/max(cnt,1); dinv = rsqrt(in_deg+1); nninv = 1/(cnt+1)
__global__ void node_scalars_kernel(const int* __restrict__ cnt, const int* __restrict__ deg,
                                    float* __restrict__ invnb, float* __restrict__ dinv,
                                    float* __restrict__ nninv, int N) {
  int i = blockIdx.x * blockDim.x + threadIdx.x;
  if (i >= N) return;
  float c = (float)cnt[i];
  invnb[i] = 1.0f / fmaxf(c, 1.0f);
  dinv[i]  = rsqrtf((float)(deg[i] + 1));   // +1 self loop, always > 0
  nninv[i] = 1.0f / (c + 1.0f);
}

// neighbor[i] = mean over CSR bucket of x[c]. One wave per node, 4 feats/lane.
// Indices batch-loaded 32 at a time and broadcast via __shfl; no float atomics.
__global__ void neighbor_gather_kernel(const float* __restrict__ x, const int* __restrict__ ofs,
                                       const int* __restrict__ adj, const float* __restrict__ invnb,
                                       float* __restrict__ neigh, int N) {
  int wave = (blockIdx.x * blockDim.x + threadIdx.x) >> 5;
  int l = threadIdx.x & 31;
  if (wave >= N) return;
  int beg = ofs[wave], end = ofs[wave + 1];
  float ax = 0.f, ay = 0.f, az = 0.f, aw = 0.f;
  for (int p0 = beg; p0 < end; p0 += 32) {
    int n32 = end - p0; if (n32 > 32) n32 = 32;
    int myc = (l < n32) ? adj[p0 + l] : 0;
    for (int j = 0; j < n32; ++j) {
      int c = __shfl(myc, j, 32);
      const float4 v = *(const float4*)(x + (size_t)c * NF + l * 4);
      ax += v.x; ay += v.y; az += v.z; aw += v.w;
    }
  }
  float s = invnb[wave];
  float4 o = { ax * s, ay * s, az * s, aw * s };
  *(float4*)(neigh + (size_t)wave * NF + l * 4) = o;
}

// hk_sum[j] = dinv[j] * sum over CSC bucket of dinv[r]*xw[r]. One wave per node.
__global__ void hk_gather_kernel(const float* __restrict__ xw, const int* __restrict__ ofs,
                                 const int* __restrict__ adj, const float* __restrict__ dinv,
                                 float* __restrict__ hk_sum, int N) {
  int wave = (blockIdx.x * blockDim.x + threadIdx.x) >> 5;
  int l = threadIdx.x & 31;
  if (wave >= N) return;
  int beg = ofs[wave], end = ofs[wave + 1];
  float ax = 0.f, ay = 0.f, az = 0.f, aw = 0.f;
  for (int p0 = beg; p0 < end; p0 += 32) {
    int n32 = end - p0; if (n32 > 32) n32 = 32;
    int myr = (l < n32) ? adj[p0 + l] : 0;
    float myw = (l < n32) ? dinv[myr] : 0.f;
    for (int j = 0; j < n32; ++j) {
      int r = __shfl(myr, j, 32);
      float w = __shfl(myw, j, 32);
      const float4 v = *(const float4*)(xw + (size_t)r * NF + l * 4);
      ax += w * v.x; ay += w * v.y; az += w * v.z; aw += w * v.w;
    }
  }
  float dj = dinv[wave];
  float4 o = { ax * dj, ay * dj, az * dj, aw * dj };
  *(float4*)(hk_sum + (size_t)wave * NF + l * 4) = o;
}

// ---------------------------------------------------------------- WMMA GEMMs
// Each wave computes a 16x16 tile of C = A[N,128] @ W[128,128]^T using
// V_WMMA_F32_16X16X4_F32, K stepped 4 at a time (32 steps).
// A layout (16x4 f32): lane<16 -> row=lh, k=4kk+{0,1}; lane>=16 -> k=4kk+{2,3}
// B layout mirrors A with columns on lanes: B[k][n] = W[n][k]
// C/D: acc[v] = C[row0 + v + 8*half][col0 + lh]

__global__ __launch_bounds__(128) void xw_gemm_kernel(const float* __restrict__ A,
                                                      const float* __restrict__ W,
                                                      float* __restrict__ C, int N) {
  int wave = blockIdx.x * 4 + (threadIdx.x >> 5);
  int lane = threadIdx.x & 31;
  int row0 = (wave >> 3) * 16;
  if (row0 >= N) return;                 // wave-uniform: EXEC stays all-ones
  int col0 = (wave & 7) * 16;
  int h = lane >> 4, lh = lane & 15;

  const float* ap = A + (size_t)(row0 + lh) * NF + h * 2;
  const float* bp = W + (size_t)(col0 + lh) * NF + h * 2;
  v8f acc = {};
#pragma unroll 8
  for (int kk = 0; kk < NF / 4; ++kk) {
    v2f a = *(const v2f*)(ap + kk * 4);
    v2f b = *(const v2f*)(bp + kk * 4);
    acc = __builtin_amdgcn_wmma_f32_16x16x4_f32(false, a, false, b, (short)0, acc, false, false);
  }
#pragma unroll
  for (int v = 0; v < 8; ++v) {
    int m = row0 + v + h * 8;
    C[(size_t)m * NF + col0 + lh] = acc[v];
  }
}

// FiLM: gamma_pre = x@g1^T + nb@g2^T ; beta_pre = x@b1^T + nb@b2^T
// out = x + (lrelu(gamma_pre)+1)*r + lrelu(beta_pre) - nb
__global__ __launch_bounds__(128) void film_kernel(const float* __restrict__ x,
                                                   const float* __restrict__ nb,
                                                   const float* __restrict__ g1,
                                                   const float* __restrict__ g2,
                                                   const float* __restrict__ b1,
                                                   const float* __restrict__ b2,
                                                   const float* __restrict__ rvec,
                                                   float* __restrict__ out, int N) {
  int wave = blockIdx.x * 4 + (threadIdx.x >> 5);
  int lane = threadIdx.x & 31;
  int row0 = (wave >> 3) * 16;
  if (row0 >= N) return;
  int col0 = (wave & 7) * 16;
  int h = lane >> 4, lh = lane & 15;

  const float* ax = x  + (size_t)(row0 + lh) * NF + h * 2;
  const float* an = nb + (size_t)(row0 + lh) * NF + h * 2;
  const float* wg1 = g1 + (size_t)(col0 + lh) * NF + h * 2;
  const float* wg2 = g2 + (size_t)(col0 + lh) * NF + h * 2;
  const float* wb1 = b1 + (size_t)(col0 + lh) * NF + h * 2;
  const float* wb2 = b2 + (size_t)(col0 + lh) * NF + h * 2;

  v8f accg = {};
  v8f accb = {};
#pragma unroll 4
  for (int kk = 0; kk < NF / 4; ++kk) {
    v2f a0 = *(const v2f*)(ax + kk * 4);
    v2f a1 = *(const v2f*)(an + kk * 4);
    v2f q0 = *(const v2f*)(wg1 + kk * 4);
    v2f q1 = *(const v2f*)(wg2 + kk * 4);
    v2f q2 = *(const v2f*)(wb1 + kk * 4);
    v2f q3 = *(const v2f*)(wb2 + kk * 4);
    accg = __builtin_amdgcn_wmma_f32_16x16x4_f32(false, a0, false, q0, (short)0, accg, false, false);
    accg = __builtin_amdgcn_wmma_f32_16x16x4_f32(false, a1, false, q1, (short)0, accg, false, false);
    accb = __builtin_amdgcn_wmma_f32_16x16x4_f32(false, a0, false, q2, (short)0, accb, false, false);
    accb = __builtin_amdgcn_wmma_f32_16x16x4_f32(false, a1, false, q3, (short)0, accb, false, false);
  }
  float rv = rvec[col0 + lh];
#pragma unroll
  for (int v = 0; v < 8; ++v) {
    int m = row0 + v + h * 8;
    size_t idx = (size_t)m * NF + col0 + lh;
    float g = accg[v]; g = (g >= 0.0f) ? g : 0.2f * g; g += 1.0f;   // lrelu(.,0.2)+1
    float b = accb[v]; b = (b >= 0.0f) ? b : 0.2f * b;
    out[idx] = x[idx] + (g * rv + b) - nb[idx];
  }
}

// h_s = output@lin^T fused with:  h_k = (hk_sum + dinv^2*xw + bias + h_s) / (cnt+1)
__global__ __launch_bounds__(128) void hs_finalize_kernel(const float* __restrict__ outp,
                                                          const float* __restrict__ W,
                                                          const float* __restrict__ hk_sum,
                                                          const float* __restrict__ xw,
                                                          const float* __restrict__ dinv,
                                                          const float* __restrict__ nninv,
                                                          const float* __restrict__ bias,
                                                          float* __restrict__ hk_out, int N) {
  int wave = blockIdx.x * 4 + (threadIdx.x >> 5);
  int lane = threadIdx.x & 31;
  int row0 = (wave >> 3) * 16;
  if (row0 >= N) return;
  int col0 = (wave & 7) * 16;
  int h = lane >> 4, lh = lane & 15;

  const float* ap = outp + (size_t)(row0 + lh) * NF + h * 2;
  const float* bp = W + (size_t)(col0 + lh) * NF + h * 2;
  v8f acc = {};
#pragma unroll 8
  for (int kk = 0; kk < NF / 4; ++kk) {
    v2f a = *(const v2f*)(ap + kk * 4);
    v2f b = *(const v2f*)(bp + kk * 4);
    acc = __builtin_amdgcn_wmma_f32_16x16x4_f32(false, a, false, b, (short)0, acc, false, false);
  }
  float bn = bias[col0 + lh];
#pragma unroll
  for (int v = 0; v < 8; ++v) {
    int m = row0 + v + h * 8;
    size_t idx = (size_t)m * NF + col0 + lh;
    float di = dinv[m];
    float hk = hk_sum[idx] + di * di * xw[idx] + bn + acc[v];
    hk_out[idx] = hk * nninv[m];
  }
}

// ---------------------------------------------------------------- launch
extern "C" void kernel_launch(void* const* d_in, const int* in_sizes, int n_in,
                              void* d_out, int out_size, void* d_ws, size_t ws_size,
                              hipStream_t stream) {
  const float* x   = (const float*)d_in[0];
  const int* ei    = (const int*)d_in[1];
  // d_in[2] = head (static 0, tail branch)
  const float* g1  = (const float*)d_in[3];
  const float* g2  = (const float*)d_in[4];
  const float* b1  = (const float*)d_in[5];
  const float* b2  = (const float*)d_in[6];
  const float* rv  = (const float*)d_in[7];
  const float* lw  = (const float*)d_in[8];
  const float* bias = (const float*)d_in[9];

  const int N = in_sizes[0] / NF;
  const int E = in_sizes[1] / 2;
  const int* row = ei;
  const int* col = ei + E;

  float* hk_out = (float*)d_out;                  // [N,128] first output
  float* outp   = (float*)d_out + (size_t)N * NF; // [N,128] second output

  // workspace layout
  float* neigh   = (float*)d_ws;                    // N*128 f
  float* xw      = neigh + (size_t)N * NF;          // N*128 f
  float* hk_sum  = xw + (size_t)N * NF;             // N*128 f
  float* dinv    = hk_sum + (size_t)N * NF;         // N
  float* nninv   = dinv + N;                        // N
  float* invnb   = nninv + N;                       // N
  int*   cnt     = (int*)(invnb + N);               // N  (zeroed)
  int*   deg     = cnt + N;                         // N  (zeroed)
  int*   row_ofs = deg + N;                         // N+1
  int*   col_ofs = row_ofs + (N + 1);               // N+1
  int*   cur_r   = col_ofs + (N + 1);               // N
  int*   cur_c   = cur_r + N;                       // N
  int*   csr_col = cur_c + N;                       // E
  int*   csc_row = csr_col + E;                     // E

  zero_int_kernel<<<(2 * N + 255) / 256, 256, 0, stream>>>(cnt, 2 * N);
  degree_kernel<<<1024, 256, 0, stream>>>(row, col, cnt, deg, E);
  scan2_kernel<<<1, 1024, 0, stream>>>(cnt, deg, row_ofs, col_ofs, N);
  fill_cursor_kernel<<<(N + 255) / 256, 256, 0, stream>>>(row_ofs, col_ofs, cur_r, cur_c, N);
  build_adj_kernel<<<1024, 256, 0, stream>>>(row, col, cur_r, cur_c, csr_col, csc_row, E);
  node_scalars_kernel<<<(N + 255) / 256, 256, 0, stream>>>(cnt, deg, invnb, dinv, nninv, N);

  {
    long long th = (long long)N * 32;
    neighbor_gather_kernel<<<(int)((th + 255) / 256), 256, 0, stream>>>(x, row_ofs, csr_col,
                                                                        invnb, neigh, N);
  }

  int waves = ((N + 15) / 16) * 8;        // one wave per 16x16 output tile
  int gblocks = (waves + 3) / 4;          // 4 waves (128 threads) per block

  film_kernel<<<gblocks, 128, 0, stream>>>(x, neigh, g1, g2, b1, b2, rv, outp, N);
  xw_gemm_kernel<<<gblocks, 128, 0, stream>>>(x, lw, xw, N);

  {
    long long th = (long long)N * 32;
    hk_gather_kernel<<<(int)((th + 255) / 256), 256, 0, stream>>>(xw, col_ofs, csc_row,
                                                                  dinv, hk_sum, N);
  }

  hs_finalize_kernel<<<gblocks, 128, 0, stream>>>(outp, lw, hk_sum, xw, dinv, nninv, bias, hk_out, N);
}